// UNet2D_4Lanes_LSF_72550587564199
// MI455X (gfx1250) — compile-verified
//
#include <hip/hip_runtime.h>

typedef __attribute__((ext_vector_type(2))) float v2f;
typedef __attribute__((ext_vector_type(8))) float v8f;

#define BATCH 32
#define CIN   3
#define COUT  4
#define HH    256
#define WW    512
#define KRED  27            // CIN*3*3 im2col reduction length
#define NCHUNK 7            // ceil(27/4) K-chunks of 4 for wmma f32 16x16x4
#define XS_ROW   (WW + 2)   // padded input row in LDS
#define XS_CINSTR (3 * XS_ROW)
#define ZROW_OFF (CIN * 3 * XS_ROW)   // extra zeroed row for k>=27 pad reads
#define OUT_STR  (WW + 4)   // padded output row (bank-conflict mitigation)

// Map im2col index k -> element offset into xs[] (tile-invariant per lane).
// k >= 27 maps into a dedicated zero row, so inner-loop loads are branch-free.
__device__ __forceinline__ int koff(int k) {
    if (k >= KRED) return ZROW_OFF;
    int cin = k / 9;
    int r   = k - 9 * cin;
    int kh  = r / 3;
    int kw  = r - 3 * kh;
    return cin * XS_CINSTR + kh * XS_ROW + kw;
}

__global__ __launch_bounds__(256) void unet_lsf_main(
    const float* __restrict__ x,      // [B,CIN,H,W]
    const float* __restrict__ masks,  // [B,COUT,H,W]
    const float* __restrict__ cw,     // [COUT,CIN,3,3] -> cw[n*27 + k]
    const float* __restrict__ cb,     // [COUT]
    float* __restrict__ weights_out,  // [B,COUT,H,W]
    float* __restrict__ acc)          // [B,COUT,2] global accumulators (pre-zeroed)
{
    __shared__ float xs[CIN * 3 * XS_ROW + XS_ROW]; // 9 padded rows + 1 zero row
    __shared__ float outw[COUT * OUT_STR];          // sigmoid weights, channel-major
    __shared__ float red[2 * COUT];                 // per-channel partial sums

    const int t    = threadIdx.x;
    const int lane = t & 31;
    const int wave = t >> 5;
    const int b    = blockIdx.x >> 8;   // / HH
    const int h    = blockIdx.x & 255;  // % HH

    // -------- phase-2 addresses, prefetch mask row early --------
    const int cph   = t >> 6;            // channel this thread reduces (t*8/512)
    const int wbase = (t & 63) * 8;      // 8 contiguous pixels
    const long rowoff = ((long)((b * COUT + cph) * HH + h) << 9) + wbase;
    const float* mrow = masks + rowoff;
    float* orow = weights_out + rowoff;
    __builtin_prefetch(mrow, 0, 1);      // -> global_prefetch_b8

    // -------- B fragments: conv weights, 4x16 per K-chunk --------
    const int n     = lane & 15;              // output channel column (only n<4 live)
    const int kbase = (lane < 16) ? 0 : 2;    // f32 wmma A/B K split across half-waves
    v2f bfrag[NCHUNK];
#pragma unroll
    for (int c = 0; c < NCHUNK; ++c) {
        int k0 = 4 * c + kbase;
        bfrag[c].x = (n < COUT && k0     < KRED) ? cw[n * KRED + k0]     : 0.0f;
        bfrag[c].y = (n < COUT && k0 + 1 < KRED) ? cw[n * KRED + k0 + 1] : 0.0f;
    }
    const float bias = (n < COUT) ? cb[n] : 0.0f;

    // -------- per-lane im2col LDS offsets, hoisted out of the tile loop ----
    int offA[NCHUNK], offB[NCHUNK];
#pragma unroll
    for (int c = 0; c < NCHUNK; ++c) {
        offA[c] = koff(4 * c + kbase);
        offB[c] = koff(4 * c + kbase + 1);
    }

    if (t < 2 * COUT) red[t] = 0.0f;

    // -------- stage 9 padded input rows into LDS --------
    for (int idx = t; idx < CIN * 3 * WW; idx += 256) {
        int col = idx & 511;
        int rr  = idx >> 9;               // cin*3 + kh, 0..8
        int cin = rr / 3;
        int kh  = rr - 3 * cin;
        int hr  = h + kh - 1;
        float v = 0.0f;
        if (hr >= 0 && hr < HH)
            v = x[((long)((b * CIN + cin) * HH + hr) << 9) + col];
        xs[cin * XS_CINSTR + kh * XS_ROW + col + 1] = v;
    }
    if (t < CIN * 3 * 2) {                // zero halo columns
        int rr = t >> 1, side = t & 1;
        int cin = rr / 3, kh = rr - 3 * cin;
        xs[cin * XS_CINSTR + kh * XS_ROW + (side ? (WW + 1) : 0)] = 0.0f;
    }
    for (int idx = t; idx < XS_ROW; idx += 256)   // zero pad row (k=27 reads)
        xs[ZROW_OFF + idx] = 0.0f;
    __syncthreads();

    // -------- conv as implicit GEMM on WMMA f32 16x16x4 --------
    // Each wave: 64 pixels = 4 tiles of 16; 7 K-chunks per tile.
    const int m = lane & 15;              // this lane's A-matrix row (pixel in tile)
#pragma unroll
    for (int tile = 0; tile < 4; ++tile) {
        const int w0 = wave * 64 + tile * 16;
        const float* xsp = xs + (w0 + m); // tile base; loads are xsp[off] only
        v8f d = {};
#pragma unroll
        for (int c = 0; c < NCHUNK; ++c) {
            v2f a;
            a.x = xsp[offA[c]];
            a.y = xsp[offB[c]];
            d = __builtin_amdgcn_wmma_f32_16x16x4_f32(
                    false, a, false, bfrag[c], (short)0, d, false, false);
        }
        // D layout: lane holds column n=lane&15; VGPR v is row M=v (+8 for upper half)
        if (n < COUT) {
            const int mofs = (lane < 16) ? 0 : 8;
#pragma unroll
            for (int v = 0; v < 8; ++v) {
                float z = d[v] + bias;
                // sigmoid via v_exp_f32 + v_rcp_f32 (avoids IEEE div expansion)
                float s = __builtin_amdgcn_rcpf(1.0f + __expf(-z));
                outw[n * OUT_STR + w0 + v + mofs] = s;
            }
        }
    }
    __syncthreads();

    // -------- coalesced store + masked reductions --------
    float4 m0 = ((const float4*)mrow)[0];
    float4 m1 = ((const float4*)mrow)[1];
    float wv[8];
#pragma unroll
    for (int e = 0; e < 8; ++e) wv[e] = outw[cph * OUT_STR + wbase + e];
    ((float4*)orow)[0] = make_float4(wv[0], wv[1], wv[2], wv[3]);
    ((float4*)orow)[1] = make_float4(wv[4], wv[5], wv[6], wv[7]);

    float mm[8] = {m0.x, m0.y, m0.z, m0.w, m1.x, m1.y, m1.z, m1.w};
    float a0 = 0.0f, a1 = 0.0f;
#pragma unroll
    for (int e = 0; e < 8; ++e) {
        float wm  = wv[e] * mm[e];
        float wm2 = wm * wm;
        a0 += wm2;
        a1 += (float)(wbase + e) * wm2;
    }
    atomicAdd(&red[cph * 2 + 0], a0);   // ds_add_f32
    atomicAdd(&red[cph * 2 + 1], a1);
    __syncthreads();

    if (t < COUT) {                     // one global atomic pair per (b,c,h)
        float hf = (float)h;
        atomicAdd(&acc[(b * COUT + t) * 2 + 0], hf * hf * red[t * 2 + 0]); // s_xx
        atomicAdd(&acc[(b * COUT + t) * 2 + 1], hf *      red[t * 2 + 1]); // s_xy
    }
}

__global__ void unet_lsf_zero(float* __restrict__ acc) {
    int t = blockIdx.x * blockDim.x + threadIdx.x;
    if (t < BATCH * COUT * 2) acc[t] = 0.0f;
}

__global__ void unet_lsf_betas(const float* __restrict__ acc,
                               float* __restrict__ betas) {
    int t = blockIdx.x * blockDim.x + threadIdx.x;
    if (t < BATCH * COUT) {
        float sxx = acc[t * 2 + 0];
        float sxy = acc[t * 2 + 1];
        betas[t] = (sxx > 0.0f) ? (sxy / sxx) : -1.0f;
    }
}

extern "C" void kernel_launch(void* const* d_in, const int* in_sizes, int n_in,
                              void* d_out, int out_size, void* d_ws, size_t ws_size,
                              hipStream_t stream) {
    const float* x     = (const float*)d_in[0];   // [32,3,256,512]
    const float* masks = (const float*)d_in[1];   // [32,4,256,512]
    const float* cw    = (const float*)d_in[2];   // [4,3,3,3]
    const float* cb    = (const float*)d_in[3];   // [4]

    float* weights_out = (float*)d_out;                                // first output
    float* betas       = (float*)d_out + (long)BATCH * COUT * HH * WW; // second
    float* acc         = (float*)d_ws;                                 // [32,4,2]

    unet_lsf_zero<<<1, 256, 0, stream>>>(acc);
    unet_lsf_main<<<BATCH * HH, 256, 0, stream>>>(x, masks, cw, cb, weights_out, acc);
    unet_lsf_betas<<<1, 128, 0, stream>>>(acc, betas);
}